// ModelNew_73315091744582
// MI455X (gfx1250) — compile-verified
//
#include <hip/hip_runtime.h>

// ---------------------------------------------------------------------------
// Fused depthwise-conv3x3 + GroupNorm + tanh/hardswish + residual + channel-LSE
// for MI455X (gfx1250). Memory-bound (~444MB HBM traffic, ~19us roofline).
// Uses TDM (tensor_load_to_lds) async tile DMA into LDS with a depth-3
// pipeline (4 buffers), s_wait_tensorcnt, native v_tanh, and non-temporal
// broadcast stores. Conv is recomputed in both passes (9 FMAs) instead of
// materializing x_conv, saving 2x134MB of HBM traffic.
// ---------------------------------------------------------------------------

#define NB   8
#define CB   64
#define HB   256
#define WB   256
#define GB   8
#define THM  4      // rows per tile, main kernel (512 blocks, deep pipeline)
#define THS  32     // rows per tile, stats kernel
#define EPSV 1e-5f

typedef unsigned int       u32;
typedef unsigned long long u64;
typedef __attribute__((ext_vector_type(4))) u32 u32x4;
typedef __attribute__((ext_vector_type(4))) int i32x4;
typedef __attribute__((ext_vector_type(8))) int i32x8;

#if defined(__HIP_DEVICE_COMPILE__) && \
    __has_builtin(__builtin_amdgcn_tensor_load_to_lds) && \
    __has_builtin(__builtin_amdgcn_s_wait_tensorcnt)
#define USE_TDM 1
#else
#define USE_TDM 0
#endif

// Decisive compile-only probe: if the device pass cannot take the TDM path,
// fail loudly instead of silently falling back to plain loads.
#if defined(__HIP_DEVICE_COMPILE__) && !USE_TDM
#error "gfx1250 TDM builtins (tensor_load_to_lds / s_wait_tensorcnt) not available"
#endif

__device__ __forceinline__ float fast_tanh(float x) {
#if defined(__HIP_DEVICE_COMPILE__) && __has_builtin(__builtin_amdgcn_tanhf)
  return __builtin_amdgcn_tanhf(x);          // native v_tanh_f32 on gfx1250
#else
  float e = __expf(2.0f * x);
  return 1.0f - 2.0f / (e + 1.0f);
#endif
}

#if USE_TDM
// Issue one TDM 2D tile load: `nrows` full-width (WB) fp32 rows starting at
// global row pointer `gsrc`, into LDS byte address `ldsByteAddr`.
// Rows past `rowsInTensor` are zero-filled by TDM OOB handling (bottom halo).
__device__ __forceinline__ void tdm_issue(const float* gsrc, u32 ldsByteAddr,
                                          int nrows, int rowsInTensor) {
  u64 ga = (u64)gsrc;
  u32x4 g0;
  g0.x = 1u;                                            // count=1 (valid D#)
  g0.y = ldsByteAddr;                                   // lds_addr [63:32]
  g0.z = (u32)ga;                                       // global_addr[31:0]
  g0.w = ((u32)(ga >> 32) & 0x01FFFFFFu) | (2u << 30);  // addr[56:32] | type=2
  i32x8 g1;
  g1[0] = (int)(2u << 16);                              // data_size=2 (4B), no mask/pad
  g1[1] = (int)(((u32)WB & 0xFFFFu) << 16);             // tensor_dim0 lo16 (=256)
  g1[2] = (int)(((u32)rowsInTensor & 0xFFFFu) << 16);   // tdim0 hi=0 | tensor_dim1 lo16
  g1[3] = (int)(((u32)WB & 0xFFFFu) << 16);             // tdim1 hi=0 | tile_dim0=256
  g1[4] = (int)((u32)nrows & 0xFFFFu);                  // tile_dim1 | tile_dim2=0
  g1[5] = WB;                                           // tensor_dim0_stride lo32
  g1[6] = 0;                                            // stride hi | dim1_stride lo16
  g1[7] = 0;
  i32x4 z4 = (i32x4)0;
#if defined(__clang_major__) && (__clang_major__ >= 23)
  i32x8 z8 = (i32x8)0;
  __builtin_amdgcn_tensor_load_to_lds(g0, g1, z4, z4, z8, 0);
#else
  __builtin_amdgcn_tensor_load_to_lds(g0, g1, z4, z4, 0);
#endif
}
#else
// Host-pass / fallback cooperative tile loader (kept so the host pass parses).
__device__ __forceinline__ void coop_load_tile(float* dst, const float* src,
                                               int nrows, int gy) {
  for (int idx = threadIdx.x; idx < nrows * WB; idx += 256) {
    int gr = gy + (idx >> 8);
    dst[idx] = (gr < HB) ? src[idx] : 0.0f;
  }
}
#endif

// 3x3 depthwise conv (cross-correlation, pad=1) at output row r (tile-local),
// column w, from an LDS tile whose row l corresponds to global row h0-1+l.
__device__ __forceinline__ float conv3x3_at(const float* buf, int r, int w,
                                            const float* wk) {
  float acc = 0.0f;
#pragma unroll
  for (int ky = 0; ky < 3; ++ky) {
    const float* row = buf + (r + ky) * WB;
    float c1 = row[w];
    float c0 = (w > 0)      ? row[w - 1] : 0.0f;
    float c2 = (w < WB - 1) ? row[w + 1] : 0.0f;
    acc = fmaf(c0, wk[ky * 3 + 0], acc);
    acc = fmaf(c1, wk[ky * 3 + 1], acc);
    acc = fmaf(c2, wk[ky * 3 + 2], acc);
  }
  return acc;
}

// ---------------------------------------------------------------------------
// Pass 1: per-(n,c,row-tile) conv statistics -> deterministic per-block partials
// grid = NB*CB*(HB/THS) = 4096 blocks, 256 threads (8 wave32)
// ---------------------------------------------------------------------------
__global__ __launch_bounds__(256) void k_stats(const float* __restrict__ x,
                                               const float* __restrict__ wgt,
                                               float* __restrict__ partials) {
  __shared__ float tile[(THS + 2) * WB];   // 34 KB
  __shared__ float redS[256];
  __shared__ float redQ[256];

  const int tid = threadIdx.x;
  const int blk = blockIdx.x;
  const int tpc = HB / THS;
  const int t   = blk % tpc;
  const int nc  = blk / tpc;               // n*CB + c
  const int c   = nc % CB;
  const int h0  = t * THS;

  float wk[9];
#pragma unroll
  for (int j = 0; j < 9; ++j) wk[j] = wgt[j * CB + c];   // weight[ky][kx][c]

  int gy, ldsOff, nrows;
  if (h0 == 0) { gy = 0;      ldsOff = 1; nrows = THS + 1; tile[tid] = 0.0f; }
  else         { gy = h0 - 1; ldsOff = 0; nrows = THS + 2; }

  const float* src = x + ((u64)nc * HB + (u64)gy) * WB;
  __syncthreads();

#if USE_TDM
  if (tid < 32) {
    tdm_issue(src, (u32)(u64)&tile[0] + (u32)(ldsOff * WB * 4), nrows, HB - gy);
    __builtin_amdgcn_s_wait_tensorcnt(0);
  }
  __syncthreads();
#else
  coop_load_tile(tile + ldsOff * WB, src, nrows, gy);
  __syncthreads();
#endif

  float sum = 0.0f, sq = 0.0f;
#pragma unroll 4
  for (int r = 0; r < THS; ++r) {
    float v = conv3x3_at(tile, r, tid, wk);
    sum += v;
    sq   = fmaf(v, v, sq);
  }
  redS[tid] = sum; redQ[tid] = sq;
  __syncthreads();
  for (int off = 128; off > 0; off >>= 1) {
    if (tid < off) { redS[tid] += redS[tid + off]; redQ[tid] += redQ[tid + off]; }
    __syncthreads();
  }
  if (tid == 0) {
    partials[2 * blk + 0] = redS[0];
    partials[2 * blk + 1] = redQ[0];
  }
}

// ---------------------------------------------------------------------------
// Pass 2: fold partials into per-(n,g) mean and rsqrt(var+eps). Deterministic.
// ---------------------------------------------------------------------------
__global__ void k_finalize(const float* __restrict__ partials,
                           float* __restrict__ meanInv) {
  int i = threadIdx.x;                      // (n,g)
  if (i >= NB * GB) return;
  int n = i / GB, g = i % GB;
  float s = 0.0f, q = 0.0f;
  for (int cc = 0; cc < CB / GB; ++cc) {
    int c = g * (CB / GB) + cc;
    for (int t = 0; t < HB / THS; ++t) {
      int blk = (n * CB + c) * (HB / THS) + t;
      s += partials[2 * blk + 0];
      q += partials[2 * blk + 1];
    }
  }
  const float cnt  = (float)((CB / GB) * HB * WB);
  float mean = s / cnt;
  float var  = q / cnt - mean * mean;
  meanInv[2 * i + 0] = mean;
  meanInv[2 * i + 1] = rsqrtf(var + EPSV);
}

// Per-channel fused accumulate: conv -> norm -> tanh -> hardswish -> residual
// -> online logsumexp update for THM rows at column w.
__device__ __forceinline__ void accum(const float* buf, const float* sw,
                                      const float* sstat, int c, int w,
                                      float* m, float* sacc) {
  float wk[9];
#pragma unroll
  for (int j = 0; j < 9; ++j) wk[j] = sw[j * CB + c];
  const int   g   = c >> 3;
  const float mu  = sstat[2 * g + 0];
  const float inv = sstat[2 * g + 1];
#pragma unroll
  for (int r = 0; r < THM; ++r) {
    float v   = conv3x3_at(buf, r, w, wk);
    float xn  = (v - mu) * inv;
    float th  = fast_tanh(xn);
    float hs  = th * fminf(fmaxf(th + 3.0f, 0.0f), 6.0f) * (1.0f / 6.0f);
    float res = v + hs;
    float nm  = fmaxf(m[r], res);
    sacc[r]   = fmaf(sacc[r], __expf(m[r] - nm), __expf(res - nm));
    m[r]      = nm;
  }
}

// ---------------------------------------------------------------------------
// Pass 3: per-(n, 4-row tile) block loops all 64 channels with a depth-3
// TDM pipeline (4 LDS buffers), computes online LSE in registers, broadcasts.
// grid = NB*(HB/THM) = 512 blocks, 256 threads.
// In-flight reads ~ 512 blocks x 3 x 6KB ~ 9MB -> hides HBM latency at 23 TB/s.
// ---------------------------------------------------------------------------
__global__ __launch_bounds__(256) void k_main(const float* __restrict__ x,
                                              const float* __restrict__ wgt,
                                              const float* __restrict__ meanInv,
                                              float* __restrict__ out) {
  __shared__ float tile[4][(THM + 2) * WB];  // 4 x 6 KB ring buffer
  __shared__ float sw[9 * CB];               // all depthwise weights (2.25 KB)
  __shared__ float sstat[2 * GB];            // mean/inv for this n

  const int tid = threadIdx.x;
  const int blk = blockIdx.x;
  const int tpn = HB / THM;
  const int n   = blk / tpn;
  const int h0  = (blk % tpn) * THM;

  for (int i = tid; i < 9 * CB; i += 256) sw[i] = wgt[i];
  if (tid < 2 * GB) sstat[tid] = meanInv[n * 2 * GB + tid];

  int gy, ldsOff, nrows;
  if (h0 == 0) {
    gy = 0; ldsOff = 1; nrows = THM + 1;
    // persistent zero halo row (row 0) in every ring buffer
    tile[0][tid] = 0.0f; tile[1][tid] = 0.0f;
    tile[2][tid] = 0.0f; tile[3][tid] = 0.0f;
  } else {
    gy = h0 - 1; ldsOff = 0; nrows = THM + 2;
  }

  const float* xn = x + (u64)n * CB * HB * WB + (u64)gy * WB;

  float m[THM], sacc[THM];
#pragma unroll
  for (int r = 0; r < THM; ++r) { m[r] = -__builtin_inff(); sacc[r] = 0.0f; }

  __syncthreads();

#if USE_TDM
  u32 ldsA[4];
#pragma unroll
  for (int b = 0; b < 4; ++b)
    ldsA[b] = (u32)(u64)&tile[b][0] + (u32)(ldsOff * WB * 4);

  if (tid < 32) {
    tdm_issue(xn,                       ldsA[0], nrows, HB - gy);
    tdm_issue(xn + (u64)1 * HB * WB,    ldsA[1], nrows, HB - gy);
    tdm_issue(xn + (u64)2 * HB * WB,    ldsA[2], nrows, HB - gy);
  }
  for (int c = 0; c < CB; ++c) {
    if (tid < 32) {
      if (c + 3 < CB) {            // steady state: keep 3 tiles in flight
        tdm_issue(xn + (u64)(c + 3) * HB * WB, ldsA[(c + 3) & 3], nrows, HB - gy);
        __builtin_amdgcn_s_wait_tensorcnt(3);   // oldest (tile c) complete
      } else if (c + 3 == CB) {
        __builtin_amdgcn_s_wait_tensorcnt(2);
      } else if (c + 2 == CB) {
        __builtin_amdgcn_s_wait_tensorcnt(1);
      } else {
        __builtin_amdgcn_s_wait_tensorcnt(0);
      }
    }
    __syncthreads();
    accum(tile[c & 3], sw, sstat, c, tid, m, sacc);
    __syncthreads();                            // before buffer is re-DMAed
  }
#else
  for (int c = 0; c < CB; ++c) {
    coop_load_tile(&tile[c & 3][ldsOff * WB], xn + (u64)c * HB * WB, nrows, gy);
    __syncthreads();
    accum(tile[c & 3], sw, sstat, c, tid, m, sacc);
    __syncthreads();
  }
#endif

  // Broadcast LSE over all channels; write-once data -> non-temporal stores.
#pragma unroll
  for (int r = 0; r < THM; ++r) {
    float lse = m[r] + __logf(sacc[r]);
    float* dst = out + ((u64)n * CB * HB + (u64)(h0 + r)) * WB + tid;
#pragma unroll 8
    for (int c = 0; c < CB; ++c)
      __builtin_nontemporal_store(lse, dst + (u64)c * HB * WB);
  }
}

// ---------------------------------------------------------------------------
extern "C" void kernel_launch(void* const* d_in, const int* in_sizes, int n_in,
                              void* d_out, int out_size, void* d_ws, size_t ws_size,
                              hipStream_t stream) {
  (void)in_sizes; (void)n_in; (void)out_size; (void)ws_size;
  const float* x   = (const float*)d_in[0];
  const float* w   = (const float*)d_in[1];
  float*       out = (float*)d_out;

  // Workspace: [0 .. 8191]  per-block partial (sum, sumsq) pairs (32 KB)
  //            [8192 .. 8319] per-(n,g) (mean, inv_std) pairs
  float* partials = (float*)d_ws;
  float* meanInv  = partials + 2 * NB * CB * (HB / THS);

  k_stats   <<<dim3(NB * CB * (HB / THS)), dim3(256), 0, stream>>>(x, w, partials);
  k_finalize<<<dim3(1),                    dim3(64),  0, stream>>>(partials, meanInv);
  k_main    <<<dim3(NB * (HB / THM)),      dim3(256), 0, stream>>>(x, w, meanInv, out);
}